// GPTPostProcess_76665166233875
// MI455X (gfx1250) — compile-verified
//
#include <hip/hip_runtime.h>
#include <hip/hip_bf16.h>
#include <stdint.h>

// GPT decode-step post-process: temperature / top-k / top-p / softmax
// B=128 rows, V=150528 vocab. Bandwidth-bound: ~77MB in + ~77MB out.
// Kernel 1: per-row histogram select (top-k kth value + nucleus cutoff + norm),
//           4-way privatized LDS histograms to break hot-bin atomic serialization.
// Kernel 2: streaming filtered softmax via async global->LDS double buffering
//           (gfx1250 GLOBAL_LOAD_ASYNC_TO_LDS_B128 + s_wait_asynccnt),
//           non-temporal B128 stores to keep the input L2-resident.

#define B_ROWS 128
#define VOCAB  150528

typedef float v4f __attribute__((ext_vector_type(4)));

#define FILTER_VALUE (-3.402823466e38f)

// ---------------------------------------------------------------------------
// gfx1250 async global->LDS copy (16B per lane), tracked by ASYNCcnt.
// VDST operand = per-lane LDS byte offset, VADDR = 64-bit global address.
__device__ __forceinline__ unsigned lds_off_of(const void* p) {
    // generic shared pointer: addr[31:0] is the workgroup-relative LDS offset
    return (unsigned)(unsigned long long)(uintptr_t)p;
}
__device__ __forceinline__ void async_copy_b128(const float* g, const void* lds_dst) {
    unsigned loff = lds_off_of(lds_dst);
    asm volatile("global_load_async_to_lds_b128 %0, %1, off"
                 :
                 : "v"(loff), "v"((unsigned long long)(uintptr_t)g)
                 : "memory");
}
#define WAIT_ASYNCCNT_0() asm volatile("s_wait_asynccnt 0" ::: "memory")
#define WAIT_ASYNCCNT_1() asm volatile("s_wait_asynccnt 1" ::: "memory")

// order-preserving key: larger float -> larger unsigned key
__device__ __forceinline__ unsigned fkey(float x) {
    unsigned u = __float_as_uint(x);
    return (u & 0x80000000u) ? ~u : (u | 0x80000000u);
}

// ---------------------------------------------------------------------------
// Kernel 1: one workgroup per row; produces {v_cut, max, 1/S2, pad} per row.
__global__ void __launch_bounds__(1024)
topk_stats_kernel(const float* __restrict__ in,
                  const float* __restrict__ temp_p,
                  const int*   __restrict__ topk_p,
                  const float* __restrict__ topp_p,
                  float*       __restrict__ params)
{
    constexpr int NB     = 4096;  // histogram bins (top 12 bits of key)
    constexpr int NSUB   = 4;     // privatized sub-histograms (contention relief)
    constexpr int CAND_N = 2048;  // candidate buffer (power of 2 for bitonic)
    constexpr int NT     = 1024;

    __shared__ unsigned hist[NSUB][NB];   // 64 KB of the 320 KB WGP LDS
    __shared__ unsigned sfx[NT];
    __shared__ float    cand[CAND_N];
    __shared__ unsigned scnt;
    __shared__ int      bselS;

    const int tid = threadIdx.x;
    const int row = blockIdx.x;
    const float temp = *temp_p;
    const float invT = (temp > 0.0f) ? (1.0f / temp) : 1.0f;
    int K = *topk_p;
    if (K < 1) K = 1;
    if (K > CAND_N) K = CAND_N;
    const float P = *topp_p;
    const float* rowp  = in + (size_t)row * VOCAB;
    const v4f*   rowp4 = (const v4f*)rowp;   // VOCAB/4 = 37632 float4s, 16B aligned

    for (int i = tid; i < NSUB * NB; i += NT) (&hist[0][0])[i] = 0u;
    if (tid == 0) { scnt = 0u; bselS = 0; }
    __syncthreads();

    // Pass 1: histogram of key top-12-bits; adjacent waves hit distinct
    // sub-histograms so Gaussian hot bins don't serialize on one address.
    const int sub = (tid >> 5) & (NSUB - 1);
    for (int i = tid; i < VOCAB / 4; i += NT) {
        v4f x = rowp4[i];
        atomicAdd(&hist[sub][fkey(x.x) >> 20], 1u);
        atomicAdd(&hist[sub][fkey(x.y) >> 20], 1u);
        atomicAdd(&hist[sub][fkey(x.z) >> 20], 1u);
        atomicAdd(&hist[sub][fkey(x.w) >> 20], 1u);
    }
    __syncthreads();

    // merge sub-histograms into hist[0]
    for (int i = tid; i < NB; i += NT)
        hist[0][i] += hist[1][i] + hist[2][i] + hist[3][i];
    __syncthreads();

    // Suffix scan over bins: find highest bucket b with count(key>=b) >= K
    unsigned part = hist[0][4 * tid] + hist[0][4 * tid + 1] +
                    hist[0][4 * tid + 2] + hist[0][4 * tid + 3];
    sfx[tid] = part;
    __syncthreads();
    for (int off = 1; off < NT; off <<= 1) {
        unsigned v = sfx[tid];
        if (tid + off < NT) v += sfx[tid + off];
        __syncthreads();
        sfx[tid] = v;
        __syncthreads();
    }
    {
        unsigned s = (tid + 1 < NT) ? sfx[tid + 1] : 0u;  // suffix above my bins
        for (int i = 3; i >= 0; --i) {
            unsigned snew = s + hist[0][4 * tid + i];
            if (snew >= (unsigned)K && s < (unsigned)K) bselS = 4 * tid + i;
            s = snew;
        }
    }
    __syncthreads();
    const unsigned bsel = (unsigned)bselS;

    // Pass 2 (L2-resident re-read): gather all elements with bucket >= bsel
    for (int i = tid; i < VOCAB / 4; i += NT) {
        v4f x = rowp4[i];
        float xs[4] = {x.x, x.y, x.z, x.w};
#pragma unroll
        for (int c = 0; c < 4; ++c) {
            if ((fkey(xs[c]) >> 20) >= bsel) {
                unsigned p = atomicAdd(&scnt, 1u);
                if (p < (unsigned)CAND_N) cand[p] = xs[c] * invT;  // store scaled
            }
        }
    }
    __syncthreads();
    const int nc = min((int)scnt, CAND_N);
    for (int i = tid; i < CAND_N; i += NT)
        if (i >= nc) cand[i] = FILTER_VALUE;
    __syncthreads();

    // Bitonic sort descending (2048 elems, 1024 threads)
    for (unsigned k = 2; k <= (unsigned)CAND_N; k <<= 1) {
        for (unsigned j = k >> 1; j > 0; j >>= 1) {
            for (unsigned t = tid; t < (unsigned)CAND_N; t += NT) {
                unsigned ixj = t ^ j;
                if (ixj > t) {
                    float a = cand[t], b = cand[ixj];
                    bool desc = ((t & k) == 0);
                    if (desc ? (a < b) : (a > b)) { cand[t] = b; cand[ixj] = a; }
                }
            }
            __syncthreads();
        }
    }

    if (tid == 0) {
        const int Kc = (K < nc) ? K : nc;
        const float m   = cand[0];
        const float kth = cand[Kc - 1];
        // top-k kept prefix + softmax sum over it (~K elements, serial is fine)
        int nk = 0; float S = 0.0f;
        while (nk < nc && cand[nk] >= kth) { S += __expf(cand[nk] - m); ++nk; }
        // nucleus: keep token j iff j==0 or cdf_{j-1} <= P (prob space)
        int nkeep; float S2;
        if (P > 0.0f) {
            float run = __expf(cand[0] - m);
            nkeep = 1;
            while (nkeep < nk && run <= P * S) {
                run += __expf(cand[nkeep] - m);
                ++nkeep;
            }
            S2 = run;                       // sum of exps over final kept set
        } else {
            nkeep = nk; S2 = S;
        }
        const float vcut = cand[nkeep - 1];
        params[row * 4 + 0] = vcut;
        params[row * 4 + 1] = m;
        params[row * 4 + 2] = 1.0f / S2;
        params[row * 4 + 3] = 0.0f;
    }
}

// ---------------------------------------------------------------------------
// Kernel 2: streaming filtered softmax. 150528 = 21 chunks * 7 iters * 256
// lanes * 4 floats. Double-buffered async global->LDS staging per lane.
#define K2_BLOCK  256
#define K2_ITERS  7
#define K2_CHUNKS 21

__global__ void __launch_bounds__(K2_BLOCK)
sample_softmax_kernel(const float* __restrict__ in,
                      const float* __restrict__ params,
                      const float* __restrict__ temp_p,
                      float*       __restrict__ out)
{
    __shared__ float buf[2][K2_BLOCK * 4];

    const int tid = threadIdx.x;
    const int row = blockIdx.y;
    const float vcut  = params[row * 4 + 0];
    const float m     = params[row * 4 + 1];
    const float invS2 = params[row * 4 + 2];
    const float temp  = *temp_p;
    const float invT  = (temp > 0.0f) ? (1.0f / temp) : 1.0f;

    const size_t chunkbase = (size_t)row * VOCAB +
                             (size_t)blockIdx.x * (K2_ITERS * K2_BLOCK * 4);

    // prime the pipeline: stage iter 0 into buf[0]
    async_copy_b128(in + chunkbase + tid * 4, &buf[0][tid * 4]);

    for (int it = 0; it < K2_ITERS; ++it) {
        const int cur = it & 1;
        if (it + 1 < K2_ITERS) {
            async_copy_b128(in + chunkbase + (size_t)(it + 1) * (K2_BLOCK * 4) + tid * 4,
                            &buf[cur ^ 1][tid * 4]);
            WAIT_ASYNCCNT_1();   // iter `it` complete (in-order), next still in flight
        } else {
            WAIT_ASYNCCNT_0();
        }

        // each lane reads back exactly the 16B it staged -> no barrier needed
        v4f x = *(const v4f*)&buf[cur][tid * 4];
        v4f o;
        {
            float t0 = x.x * invT, t1 = x.y * invT, t2 = x.z * invT, t3 = x.w * invT;
            o.x = (t0 >= vcut) ? __expf(t0 - m) * invS2 : 0.0f;
            o.y = (t1 >= vcut) ? __expf(t1 - m) * invS2 : 0.0f;
            o.z = (t2 >= vcut) ? __expf(t2 - m) * invS2 : 0.0f;
            o.w = (t3 >= vcut) ? __expf(t3 - m) * invS2 : 0.0f;
        }
        // non-temporal: keep the write stream out of L2 (input stays resident)
        __builtin_nontemporal_store(
            o, (v4f*)(out + chunkbase + (size_t)it * (K2_BLOCK * 4) + tid * 4));
    }
}

// ---------------------------------------------------------------------------
extern "C" void kernel_launch(void* const* d_in, const int* in_sizes, int n_in,
                              void* d_out, int out_size, void* d_ws, size_t ws_size,
                              hipStream_t stream) {
    const float* in     = (const float*)d_in[0];
    // d_in[1] = batch_seqlen (unused for decode path)
    const float* temp_p = (const float*)d_in[2];
    const int*   topk_p = (const int*)d_in[3];
    const float* topp_p = (const float*)d_in[4];
    float* params = (float*)d_ws;   // B_ROWS * 4 floats
    float* out    = (float*)d_out;

    topk_stats_kernel<<<B_ROWS, 1024, 0, stream>>>(in, temp_p, topk_p, topp_p, params);

    dim3 g2(K2_CHUNKS, B_ROWS);
    sample_softmax_kernel<<<g2, K2_BLOCK, 0, stream>>>(in, params, temp_p, out);
}